// TestPoincare_28467043237896
// MI455X (gfx1250) — compile-verified
//
#include <hip/hip_runtime.h>

typedef float v2f __attribute__((ext_vector_type(2)));
typedef float v8f __attribute__((ext_vector_type(8)));

#define DIMK 64
#define KSTEPS 16  // DIMK / 4

// Load one 16x(K=64) fp32 fragment for V_WMMA_F32_16X16X4_F32.
// Layout per k-step s (abs K = 4s..4s+3):
//   lanes 0-15  : row = lane,    elems (4s+0, 4s+1)
//   lanes 16-31 : row = lane-16, elems (4s+2, 4s+3)
// Also produces the full squared row norm in every lane (via xor-16 reduce).
__device__ __forceinline__ void load_frag64(const float* __restrict__ base,
                                            int lane, v2f frag[KSTEPS],
                                            float& rownorm2) {
    const int row  = lane & 15;
    const int koff = (lane >> 4) * 2;
    const float* p = base + (size_t)row * DIMK + koff;
    float s = 0.0f;
#pragma unroll
    for (int k = 0; k < KSTEPS; ++k) {
        v2f v = *(const v2f*)(p + 4 * k);   // global_load_b64
        frag[k] = v;
        s = fmaf(v.x, v.x, s);
        s = fmaf(v.y, v.y, s);
    }
    rownorm2 = s + __shfl_xor(s, 16, 32);   // combine lane halves -> full ||row||^2
}

__global__ __launch_bounds__(256)
void poincare_wmma_f32_kernel(const float* __restrict__ A,
                              const float* __restrict__ B,
                              float* __restrict__ out,
                              int n, int m) {
    const int lane = threadIdx.x & 31;
    const int wave = threadIdx.x >> 5;           // 0..7
    const int wi   = wave >> 1;                  // 0..3  (i direction)
    const int wj   = wave & 1;                   // 0..1  (j direction)

    const int i0 = blockIdx.y * 128 + wi * 32;   // 32 rows of a per wave
    const int j0 = blockIdx.x * 64  + wj * 32;   // 32 rows of b per wave

    // ---- load fragments (stay in registers; inputs are L2-resident) ----
    v2f afr[2][KSTEPS];  float a2[2];
    v2f bfr[2][KSTEPS];  float b2[2];
    load_frag64(A + (size_t)i0 * DIMK,        lane, afr[0], a2[0]);
    load_frag64(A + (size_t)(i0 + 16) * DIMK, lane, afr[1], a2[1]);
    load_frag64(B + (size_t)j0 * DIMK,        lane, bfr[0], b2[0]);
    load_frag64(B + (size_t)(j0 + 16) * DIMK, lane, bfr[1], b2[1]);

    // Per-row / per-col reciprocal factors: rcp((1-a2)(1-b2)) = rcp(1-a2)*rcp(1-b2).
    // Computed once per fragment instead of once per output element.
    float ra[2], rb[2];
#pragma unroll
    for (int t = 0; t < 2; ++t) {
        ra[t] = __builtin_amdgcn_rcpf(1.0f - a2[t]);
        rb[t] = __builtin_amdgcn_rcpf(1.0f - b2[t]);
    }

    // ---- 2x2 tiles of 16x16, K=64 via 16 x V_WMMA_F32_16X16X4_F32 each ----
    v8f c[2][2] = {};
#pragma unroll
    for (int k = 0; k < KSTEPS; ++k) {
#pragma unroll
        for (int ti = 0; ti < 2; ++ti) {
#pragma unroll
            for (int tj = 0; tj < 2; ++tj) {
                c[ti][tj] = __builtin_amdgcn_wmma_f32_16x16x4_f32(
                    /*neg_a=*/false, afr[ti][k],
                    /*neg_b=*/false, bfr[tj][k],
                    /*c_mod=*/(short)0, c[ti][tj],
                    /*reuse_a=*/false, /*reuse_b=*/false);
            }
        }
    }

    // ---- fused Poincare epilogue + non-temporal store ----
    // C/D layout: lane ll (col = ll&15); VGPR v -> row v (lanes 0-15) or v+8 (lanes 16-31)
    const int  col = lane & 15;
    const bool hi  = lane >= 16;

#pragma unroll
    for (int ti = 0; ti < 2; ++ti) {
#pragma unroll
        for (int tj = 0; tj < 2; ++tj) {
            const float b2v  = b2[tj];          // lane already holds ||b_(j0+col)||^2
            const float rb2v = 2.0f * rb[tj];   // 2 * rcp(1 - b2)
#pragma unroll
            for (int v = 0; v < 8; ++v) {
                const float a2lo = __shfl(a2[ti], v, 32);       // row v
                const float a2hi = __shfl(a2[ti], v + 8, 32);   // row v+8
                const float ralo = __shfl(ra[ti], v, 32);
                const float rahi = __shfl(ra[ti], v + 8, 32);
                const float a2v  = hi ? a2hi : a2lo;
                const float rav  = hi ? rahi : ralo;

                const float dot = c[ti][tj][v];
                float sq = a2v + b2v - 2.0f * dot;
                sq = fmaxf(sq, 0.0f);
                const float dist  = __builtin_amdgcn_sqrtf(sq);
                // inner = 1 + 2*d / ((1-a2)(1-b2)) = 1 + d*rav*rb2v
                const float inner = fmaf(dist * rav, rb2v, 1.0f);
                const float t = inner +
                    __builtin_amdgcn_sqrtf(fmaf(inner, inner, -1.0f));
                const float res = __logf(t);   // arcosh(inner)

                const int row = i0 + ti * 16 + v + (hi ? 8 : 0);
                const int cg  = j0 + tj * 16 + col;
                __builtin_nontemporal_store(res, &out[(size_t)row * m + cg]);
            }
        }
    }
    (void)n;
}

extern "C" void kernel_launch(void* const* d_in, const int* in_sizes, int n_in,
                              void* d_out, int out_size, void* d_ws, size_t ws_size,
                              hipStream_t stream) {
    const float* a = (const float*)d_in[0];   // [n, 64]
    const float* b = (const float*)d_in[1];   // [m, 64]
    float* out = (float*)d_out;               // [n, m]

    const int n = in_sizes[0] / DIMK;         // 8192
    const int m = in_sizes[1] / DIMK;         // 8192

    dim3 grid((m + 63) / 64, (n + 127) / 128);
    dim3 block(256, 1, 1);
    poincare_wmma_f32_kernel<<<grid, block, 0, stream>>>(a, b, out, n, m);

    (void)n_in; (void)out_size; (void)d_ws; (void)ws_size;
}